// KernelQM9_86414741995618
// MI455X (gfx1250) — compile-verified
//
#include <hip/hip_runtime.h>
#include <math.h>

typedef float v2f __attribute__((ext_vector_type(2)));
typedef float v8f __attribute__((ext_vector_type(8)));
typedef float f4v __attribute__((ext_vector_type(4)));

#define NPTS   16      // points per workgroup (== WMMA M)
#define NPATH  1536    // radial output channels per point
#define RH     64      // radial hidden width (== WMMA K total)
#define NDIM   64

// silu
__device__ __forceinline__ float siluf(float x) { return x / (1.0f + __expf(-x)); }

__global__ __launch_bounds__(256, 1) void KernelQM9_fused(
    const float* __restrict__ rin,   // (16384, 3)
    const float* __restrict__ W1,    // (1, 64)
    const float* __restrict__ b1,    // (64)
    const float* __restrict__ W2,    // (64, 1536) row-major
    const float* __restrict__ b2,    // (1536)
    float* __restrict__ out)         // (16384, 64, 64)
{
    extern __shared__ float smem[];
    float* Rlds = smem;                      // 16*1536
    float* hlds = Rlds + NPTS * NPATH;       // 16*64
    float* rad  = hlds + NPTS * RH;          // 16
    float* A00  = rad  + NPTS;               // 16
    float* A01  = A00  + NPTS;               // 16*3
    float* A10  = A01  + NPTS * 3;           // 16*3
    float* CM   = A10  + NPTS * 3;           // 16*27  [p][f][i][j]

    const int tid = threadIdx.x;
    const long long pbase = (long long)blockIdx.x * NPTS;

    // ---------------- Phase A: per-point SH + folded CG/norm coefficients ----
    if (tid < NPTS) {
        const float x = rin[(pbase + tid) * 3 + 0];
        const float y = rin[(pbase + tid) * 3 + 1];
        const float z = rin[(pbase + tid) * 3 + 2];
        const float r2 = x * x + y * y + z * z;
        const bool  zr = (r2 == 0.0f);
        const float inv  = rsqrtf(r2);
        const float inv2 = 1.0f / r2;

        const float SH_C0  = 0.28209479177387814f;   // 0.5/sqrt(pi)
        const float SH_C1  = 0.48860251190291992f;   // sqrt(3/(4pi))
        const float SH_C2  = 1.09254843059207907f;   // sqrt(15/(4pi))
        const float SH_C20 = 0.31539156525252001f;   // sqrt(5/(16pi))

        const float Y0 = SH_C0;
        const float Yy = SH_C1 * y * inv;
        const float Yz = SH_C1 * z * inv;
        const float Yx = SH_C1 * x * inv;
        const float q0 = SH_C2 * x * y * inv2;                  // xy
        const float q1 = SH_C2 * y * z * inv2;                  // yz
        const float q2 = SH_C20 * (3.0f * z * z - r2) * inv2;   // 3z^2-r^2
        const float q3 = SH_C2 * x * z * inv2;                  // xz
        const float q4 = 0.5f * SH_C2 * (x * x - y * y) * inv2; // x^2-y^2

        rad[tid] = sqrtf(r2);

        // NORM[i][j][zero?]: lm = sqrt(4pi(2li+1)); /sqrt(nse) or /sqrt(mi)
        const float snc00 = zr ? 0.88622693f : 0.62665707f;  // sqrt(4pi)/{4,sqrt32}
        const float snc01 = zr ? 1.53499006f : 1.08544276f;  // sqrt(12pi)/{4,sqrt32}
        const float snc10 = zr ? 0.88622693f : 0.44311346f;  // sqrt(4pi)/{4,8}
        const float snc11 = zr ? 1.53499006f : 0.76749503f;  // sqrt(12pi)/{4,8}

        const float INV_SQRT3 = 0.57735026919f;
        const float INV_SQRT6 = 0.40824829046f;
        const float INV_SQRT10 = 0.31622776601f;
        const float INV_SQRT30 = 0.18257418584f;

        A00[tid] = Y0 * snc00;                 // CG(0,0,0) = 1
        const float t01 = INV_SQRT3 * snc01;   // CG(0,1,1) = delta/sqrt3
        A01[tid * 3 + 0] = t01 * Yy;
        A01[tid * 3 + 1] = t01 * Yz;
        A01[tid * 3 + 2] = t01 * Yx;
        const float t10 = INV_SQRT3 * snc10;   // CG(1,0,1) = delta/sqrt3
        A10[tid * 3 + 0] = t10 * Yy;
        A10[tid * 3 + 1] = t10 * Yz;
        A10[tid * 3 + 2] = t10 * Yx;

        float* cm = CM + tid * 27;
        // filter l=0 : delta_ij * Y0 / sqrt3
        const float m0 = INV_SQRT3 * Y0 * snc11;
        cm[0] = m0; cm[1] = 0.f; cm[2] = 0.f;
        cm[3] = 0.f; cm[4] = m0; cm[5] = 0.f;
        cm[6] = 0.f; cm[7] = 0.f; cm[8] = m0;
        // filter l=1 : eps_ijk Y1[k] / sqrt6  (basis order y,z,x)
        const float w1s = INV_SQRT6 * snc11;
        float* e = cm + 9;
        e[0] = 0.f;        e[1] =  w1s * Yx;  e[2] = -w1s * Yz;
        e[3] = -w1s * Yx;  e[4] = 0.f;        e[5] =  w1s * Yy;
        e[6] =  w1s * Yz;  e[7] = -w1s * Yy;  e[8] = 0.f;
        // filter l=2 : symmetric traceless projector / sqrt5
        const float s10 = INV_SQRT10 * snc11;
        const float s30 = INV_SQRT30 * snc11;
        float* s = cm + 18;
        s[0] = -s30 * q2 - s10 * q4;         // (y,y)
        s[4] = 2.0f * s30 * q2;              // (z,z)
        s[8] = -s30 * q2 + s10 * q4;         // (x,x)
        s[1] = s10 * q1;  s[3] = s[1];       // (y,z)
        s[5] = s10 * q3;  s[7] = s[5];       // (z,x)
        s[2] = s10 * q0;  s[6] = s[2];       // (y,x)
    }
    __syncthreads();

    // ---------------- radial hidden layer: h[p][c] = silu(rad[p]*W1[c]+b1[c])
    #pragma unroll
    for (int q = 0; q < 4; ++q) {
        const int e = tid + 256 * q;         // 0..1023
        const int p = e >> 6;
        const int c = e & 63;
        hlds[p * RH + c] = siluf(rad[p] * W1[c] + b1[c]);
    }
    __syncthreads();

    // ---------------- Phase B: WMMA f32 GEMM  R[16x1536] = h[16x64] @ W2 + b2
    {
        const int wave = tid >> 5;
        const int lane = tid & 31;
        const int m16  = lane & 15;
        const int khalf = (lane >> 4) << 1;  // 0 (lanes 0-15) or 2 (lanes 16-31)

        // Hoist A fragments: 16 k-steps, 2 VGPRs each (K=0/1 low half, K=2/3 high)
        v2f afrag[16];
        #pragma unroll
        for (int kk = 0; kk < 16; ++kk) {
            const int kb = 4 * kk + khalf;
            afrag[kk].x = hlds[m16 * RH + kb];
            afrag[kk].y = hlds[m16 * RH + kb + 1];
        }

        for (int t = 0; t < 12; ++t) {       // 8 waves * 12 tiles = 96 * 16 = 1536
            const int n0 = (wave * 12 + t) * 16;
            v8f acc = {};
            #pragma unroll
            for (int kk = 0; kk < 16; ++kk) {
                const int kb = 4 * kk + khalf;
                v2f bfrag;
                bfrag.x = W2[kb * NPATH + n0 + m16];
                bfrag.y = W2[(kb + 1) * NPATH + n0 + m16];
                acc = __builtin_amdgcn_wmma_f32_16x16x4_f32(
                    false, afrag[kk], false, bfrag, (short)0, acc, false, false);
            }
            const float bias = b2[n0 + m16];
            const int mbase = (lane < 16) ? 0 : 8;
            #pragma unroll
            for (int g = 0; g < 8; ++g)
                Rlds[(mbase + g) * NPATH + n0 + m16] = acc[g] + bias;
        }
    }
    __syncthreads();

    // ---------------- Phase C: assemble 64x64 kernel per point, stream out ---
    {
        const int pp    = tid >> 4;          // point 0..15
        const int cbase = (tid & 15) * 4;    // 4 consecutive output cols
        const float* Rp = Rlds + pp * NPATH;
        float* outp = out + (pbase + pp) * (NDIM * NDIM) + cbase;

        const float a00 = A00[pp];
        const bool colL0 = (cbase < 16);
        int vj_v[4], vj_j[4];
        #pragma unroll
        for (int c = 0; c < 4; ++c) {
            const int cc = cbase + c - 16;   // valid only when !colL0
            vj_v[c] = (cc >= 0) ? cc / 3 : 0;
            vj_j[c] = (cc >= 0) ? cc - vj_v[c] * 3 : 0;
        }

        #pragma unroll 4
        for (int row = 0; row < NDIM; ++row) {
            f4v v;
            if (row < 16) {
                const int u = row;
                if (colL0) {
                    #pragma unroll
                    for (int c = 0; c < 4; ++c)
                        v[c] = a00 * Rp[u * 16 + cbase + c];
                } else {
                    #pragma unroll
                    for (int c = 0; c < 4; ++c)
                        v[c] = A01[pp * 3 + vj_j[c]] * Rp[256 + u * 16 + vj_v[c]];
                }
            } else {
                const int rw = row - 16;
                const int u = rw / 3;
                const int i = rw - u * 3;
                if (colL0) {
                    const float a10 = A10[pp * 3 + i];
                    #pragma unroll
                    for (int c = 0; c < 4; ++c)
                        v[c] = a10 * Rp[512 + u * 16 + cbase + c];
                } else {
                    #pragma unroll
                    for (int c = 0; c < 4; ++c) {
                        const float* rws = Rp + 768 + (u * 16 + vj_v[c]) * 3;
                        const float* cmp = CM + pp * 27 + i * 3 + vj_j[c];
                        v[c] = cmp[0] * rws[0] + cmp[9] * rws[1] + cmp[18] * rws[2];
                    }
                }
            }
            __builtin_nontemporal_store(v, (f4v*)(outp + row * NDIM));
        }
    }
}

extern "C" void kernel_launch(void* const* d_in, const int* in_sizes, int n_in,
                              void* d_out, int out_size, void* d_ws, size_t ws_size,
                              hipStream_t stream) {
    const float* r  = (const float*)d_in[0];
    const float* W1 = (const float*)d_in[1];
    const float* b1 = (const float*)d_in[2];
    const float* W2 = (const float*)d_in[3];
    const float* b2 = (const float*)d_in[4];
    float* out = (float*)d_out;

    const int npts = in_sizes[0] / 3;        // 16384
    const int nblocks = npts / NPTS;         // 1024

    const size_t shmem =
        (size_t)(NPTS * NPATH + NPTS * RH + NPTS + NPTS + NPTS * 3 + NPTS * 3 + NPTS * 27)
        * sizeof(float);                     // ~102 KB dynamic LDS

    (void)hipFuncSetAttribute((const void*)KernelQM9_fused,
                              hipFuncAttributeMaxDynamicSharedMemorySize,
                              (int)shmem);

    KernelQM9_fused<<<nblocks, 256, shmem, stream>>>(r, W1, b1, W2, b2, out);
}